// UncertaintyCalculator_35210141893376
// MI455X (gfx1250) — compile-verified
//
#include <hip/hip_runtime.h>

// Problem constants (match reference)
#define B_  64
#define C_  256
#define HW_ 1024   // 32*32
#define K_  20
#define D_  2048

typedef __attribute__((ext_vector_type(2))) float v2f;
typedef __attribute__((ext_vector_type(8))) float v8f;

// ---------------------------------------------------------------------------
// Kernel 1: adaptive_avg_pool2d -> z[b,c] = mean over 32x32 spatial
// grid = B*C blocks, 256 threads; each thread loads one float4 (coalesced).
// ---------------------------------------------------------------------------
__global__ __launch_bounds__(256) void pool_kernel(const float* __restrict__ feat,
                                                   float* __restrict__ z) {
    const int row = blockIdx.x;                    // row = b*C + c
    const float4* p = reinterpret_cast<const float4*>(feat + (size_t)row * HW_);
    float4 v = p[threadIdx.x];
    float s = v.x + v.y + v.z + v.w;

    __shared__ float red[256];
    red[threadIdx.x] = s;
    __syncthreads();
    for (int off = 128; off > 0; off >>= 1) {
        if (threadIdx.x < off) red[threadIdx.x] += red[threadIdx.x + off];
        __syncthreads();
    }
    if (threadIdx.x == 0) z[row] = red[0] * (1.0f / (float)HW_);
}

// ---------------------------------------------------------------------------
// Kernel 2: p2[c] = sum_d P[d,c]^2   (column sums of projection matrix)
// grid = 16 blocks x 256 threads; block j sweeps 128 rows, fully coalesced,
// accumulates partials with a float atomic (p2 pre-zeroed by memsetAsync).
// ---------------------------------------------------------------------------
__global__ __launch_bounds__(256) void colnorm2_kernel(const float* __restrict__ P,
                                                       float* __restrict__ p2) {
    const int c  = threadIdx.x;
    const int d0 = blockIdx.x * (D_ / 16);
    float s = 0.f;
    for (int d = d0; d < d0 + (D_ / 16); ++d) {
        float v = P[(size_t)d * C_ + c];
        s += v * v;
    }
    atomicAdd(&p2[c], s);
}

// ---------------------------------------------------------------------------
// Kernel 3: M[k,c] = sum_b labels[b,k] * z[b,c]  via V_WMMA_F32_16X16X4_F32
// One wave (32 threads) per 16x16 output tile. 2 class tiles x 16 channel
// tiles = 32 blocks. Contraction over batch: 16 WMMA steps of K=4.
// ISA layouts (wave32):
//   A 16x4 : lane L -> row M = L%16 ; VGPR v, half h=L/16 -> K = h*2+v
//   B 4x16 : lane L -> col N = L%16 ; VGPR v, half h      -> K = h*2+v
//   D 16x16: VGPR r, lane L -> row M = r + 8*(L/16), col N = L%16
// ---------------------------------------------------------------------------
__global__ __launch_bounds__(32) void protoM_kernel(const int* __restrict__ labels,
                                                    const float* __restrict__ z,
                                                    float* __restrict__ Mpad) {
    const int lane  = threadIdx.x;        // 0..31, one full wave, EXEC all ones
    const int mtile = blockIdx.x & 1;     // class tile (0: k=0..15, 1: k=16..31)
    const int ctile = blockIdx.x >> 1;    // channel tile 0..15
    const int half  = lane >> 4;          // 0 or 1
    const int lm    = lane & 15;

    const int k_cls = mtile * 16 + lm;              // A row (class index, may pad)
    const int kc    = (k_cls < K_) ? k_cls : 0;     // clamped address
    const float kmask = (k_cls < K_) ? 1.0f : 0.0f; // zero the pad rows
    const int n = ctile * 16 + lm;                  // B column (channel)

    v8f acc = {};
    for (int s = 0; s < 16; ++s) {
        const int b0 = s * 4 + half * 2;   // batches for vgpr0/vgpr1 of this lane
        v2f a, b;
        a.x = kmask * (float)labels[(b0    ) * K_ + kc];
        a.y = kmask * (float)labels[(b0 + 1) * K_ + kc];
        b.x = z[(size_t)(b0    ) * C_ + n];
        b.y = z[(size_t)(b0 + 1) * C_ + n];
        acc = __builtin_amdgcn_wmma_f32_16x16x4_f32(
            /*neg_a=*/false, a, /*neg_b=*/false, b,
            /*c_mod=*/(short)0, acc, /*reuse_a=*/false, /*reuse_b=*/false);
    }

    // Store padded M (32 x 256); rows >= 20 are garbage-zero and never read.
    #pragma unroll
    for (int r = 0; r < 8; ++r) {
        const int kout = mtile * 16 + r + 8 * half;
        Mpad[(size_t)kout * C_ + ctile * 16 + lm] = acc[r];
    }
}

// ---------------------------------------------------------------------------
// Kernel 4: closed-form sim, masked max over k, mean over c, out = 1 - mean.
// grid = B blocks x 256 threads (thread = channel).
// ---------------------------------------------------------------------------
__global__ __launch_bounds__(256) void finalize_kernel(const float* __restrict__ z,
                                                       const float* __restrict__ p2v,
                                                       const float* __restrict__ Mpad,
                                                       const int* __restrict__ labels,
                                                       float* __restrict__ out) {
    const int b = blockIdx.x;
    const int c = threadIdx.x;

    const float a  = z[(size_t)b * C_ + c];
    const float p2 = p2v[c];
    const float hn = fmaxf(sqrtf(a * a * p2), 1e-8f);   // ||hyper|| over D

    float maxsim = -INFINITY;
    #pragma unroll
    for (int k = 0; k < K_; ++k) {
        const float m  = Mpad[(size_t)k * C_ + c];
        const float bq = sqrtf(m * m * p2);             // ||bundle|| over D
        const float t  = fmaxf(bq, 1e-12f);
        const float num = a * m * p2 / t;               // <hyper, proto> over D
        const float pn  = fmaxf(bq / t, 1e-8f);         // ||proto|| over D
        const float sim = num / (hn * pn);
        const float ms  = (labels[b * K_ + k] != 0) ? sim : -1.0f;
        maxsim = fmaxf(maxsim, ms);
    }

    __shared__ float red[256];
    red[c] = maxsim;
    __syncthreads();
    for (int off = 128; off > 0; off >>= 1) {
        if (c < off) red[c] += red[c + off];
        __syncthreads();
    }
    if (c == 0) out[b] = 1.0f - red[0] * (1.0f / (float)C_);
}

// ---------------------------------------------------------------------------
// Launch
// ---------------------------------------------------------------------------
extern "C" void kernel_launch(void* const* d_in, const int* in_sizes, int n_in,
                              void* d_out, int out_size, void* d_ws, size_t ws_size,
                              hipStream_t stream) {
    const float* feat   = (const float*)d_in[0];   // [B, C, 32, 32]
    const int*   labels = (const int*)  d_in[1];   // [B, K]
    const float* proj   = (const float*)d_in[2];   // [D, C]
    float*       out    = (float*)d_out;           // [B]

    float* ws   = (float*)d_ws;
    float* z    = ws;                 // B*C      = 16384 floats
    float* p2   = ws + B_ * C_;       // C        = 256 floats
    float* Mpad = p2 + C_;            // 32*C     = 8192 floats (padded classes)

    // zero the atomic accumulator (capture-safe async memset node)
    hipMemsetAsync(p2, 0, C_ * sizeof(float), stream);

    pool_kernel<<<B_ * C_, 256, 0, stream>>>(feat, z);
    colnorm2_kernel<<<16, 256, 0, stream>>>(proj, p2);
    protoM_kernel<<<32, 32, 0, stream>>>(labels, z, Mpad);
    finalize_kernel<<<B_, 256, 0, stream>>>(z, p2, Mpad, labels, out);
}